// BECEModel_38912403702282
// MI455X (gfx1250) — compile-verified
//
#include <hip/hip_runtime.h>
#include <hip/hip_bf16.h>
#include <math.h>

// ---------------------------------------------------------------------------
// MI455X (gfx1250) implementation.
//   * Matrix engine: V_WMMA_F32_16X16X4_F32 (wave32, fp32 A/B/C) -> reference
//     fp32 numerics on the WMMA pipe.
//   * Data movement: GLOBAL_LOAD_ASYNC_TO_LDS_B128 for LDS staging/gathers
//     (ASYNCcnt-tracked, bypasses VGPRs), explicit s_wait_asynccnt.
//   * M=32 tiles: two A tiles share each B fragment -> halves B vmem traffic
//     per WMMA.
//
// Fragment layouts (CDNA5 ISA §7.12.2, wave32):
//   A (16x4 f32):  lane l, vgpr v -> M = l&15,        K = v + 2*(l>>4)
//   B (4x16 f32):  lane l, vgpr v -> N = l&15,        K = v + 2*(l>>4)
//   C/D (16x16):   lane l, vgpr v -> M = v + 8*(l>>4), N = l&15
// ---------------------------------------------------------------------------

typedef __attribute__((ext_vector_type(2))) float v2f;
typedef __attribute__((ext_vector_type(8))) float v8f;

__device__ __forceinline__ v8f wmma4(v2f a, v2f b, v8f c) {
  // 8 args: (neg_a, A, neg_b, B, c_mod, C, reuse_a, reuse_b)
  return __builtin_amdgcn_wmma_f32_16x16x4_f32(false, a, false, b, (short)0, c,
                                               false, false);
}

#if defined(__HIP_DEVICE_COMPILE__) && defined(__gfx1250__)
#define USE_ASYNC 1
#else
#define USE_ASYNC 0
#endif

// Copy 16 bytes global -> LDS on the async pipe (per-lane addresses, so it
// also works as a gather). LDS byte address = low 32 bits of the generic
// pointer.
__device__ __forceinline__ void async_copy_b128(const float* __restrict__ g,
                                                float* l) {
#if USE_ASYNC
  unsigned int laddr = (unsigned int)(unsigned long long)l;
  unsigned long long gaddr = (unsigned long long)g;
  asm volatile("global_load_async_to_lds_b128 %0, %1, off"
               :
               : "v"(laddr), "v"(gaddr)
               : "memory");
#else
  *(float4*)l = *(const float4*)g;
#endif
}

__device__ __forceinline__ void async_join() {
#if USE_ASYNC
#if __has_builtin(__builtin_amdgcn_s_wait_asynccnt)
  __builtin_amdgcn_s_wait_asynccnt(0);
#else
  asm volatile("s_wait_asynccnt 0x0" ::: "memory");
#endif
#endif
}

#define HD 64                 // hidden dim
#define XCHUNK 256            // LDS staging chunk (K direction)
#define XSTRIDE (XCHUNK + 4)  // 260 % 64 = 4 -> conflict-free A loads
#define HSTRIDE (256 + 4)     // hidden concat row stride
#define ESTRIDE (128 + 4)     // edge raw/emb row stride
#define MSTRIDE (64 + 4)      // m1 row stride

// ---------------------------------------------------------------------------
// Kernel 1: node encoder. 32 nodes per block, 128 threads = 4 waves.
// Wave w computes output-column slice [16w,16w+16) of every sub-matmul for
// BOTH 16-row tiles; per-wave K-work = 512+768+16+8 = 1304 (perfect balance).
// ---------------------------------------------------------------------------
__device__ __forceinline__ void encode_feature(
    const float* __restrict__ X, const float* __restrict__ W,
    const float* __restrict__ B, int K, int f, float* lds_x, float* lds_h,
    int node0, int N, int tid, int wave, int row, int hi) {
  v8f acc0 = {}, acc1 = {};
  const int col = 16 * wave + row;
  for (int kbase = 0; kbase < K; kbase += XCHUNK) {
    const int ck = (K - kbase < XCHUNK) ? (K - kbase) : XCHUNK;
    const int nv = ck >> 2;  // float4 per row
    // async stage of the [32, ck] input tile (rows clamped: clamped rows only
    // feed output rows that are never stored)
    for (int idx = tid; idx < 32 * nv; idx += 128) {
      const int r = idx / nv;
      const int c4 = idx - r * nv;
      int rr = node0 + r;
      rr = (rr < N) ? rr : (N - 1);
      async_copy_b128(X + (size_t)rr * K + kbase + 4 * c4,
                      lds_x + r * XSTRIDE + 4 * c4);
    }
    async_join();
    __syncthreads();
    for (int k = 0; k < ck; k += 4) {
      v2f b;  // one B fragment shared by both row tiles
      b.x = W[(size_t)(kbase + k + 2 * hi) * HD + col];
      b.y = W[(size_t)(kbase + k + 2 * hi + 1) * HD + col];
      v2f a0 = *(const v2f*)(lds_x + row * XSTRIDE + k + 2 * hi);
      v2f a1 = *(const v2f*)(lds_x + (16 + row) * XSTRIDE + k + 2 * hi);
      acc0 = wmma4(a0, b, acc0);
      acc1 = wmma4(a1, b, acc1);
    }
    __syncthreads();  // lds_x reused by the next chunk/feature
  }
  const float bias = B[col];
  for (int r8 = 0; r8 < 8; ++r8) {
    lds_h[(r8 + 8 * hi) * HSTRIDE + f * HD + col] =
        fmaxf(acc0[r8] + bias, 0.f);
    lds_h[(16 + r8 + 8 * hi) * HSTRIDE + f * HD + col] =
        fmaxf(acc1[r8] + bias, 0.f);
  }
}

__global__ __launch_bounds__(128) void node_encode_kernel(
    const float* __restrict__ image, const float* __restrict__ tweets,
    const float* __restrict__ nump, const float* __restrict__ cat,
    const float* __restrict__ Wi, const float* __restrict__ bi,
    const float* __restrict__ Wt, const float* __restrict__ bt,
    const float* __restrict__ Wn, const float* __restrict__ bn,
    const float* __restrict__ Wc, const float* __restrict__ bc,
    const float* __restrict__ Wf, const float* __restrict__ bf,
    float* __restrict__ nf, int N) {
  __shared__ float lds_x[32 * XSTRIDE];  // ~33 KB
  __shared__ float lds_h[32 * HSTRIDE];  // ~33 KB
  const int tid = threadIdx.x;
  const int wave = tid >> 5;
  const int lane = tid & 31;
  const int row = lane & 15;
  const int hi = lane >> 4;
  const int node0 = blockIdx.x * 32;

  encode_feature(image,  Wi, bi, 512, 0, lds_x, lds_h, node0, N, tid, wave, row, hi);
  encode_feature(tweets, Wt, bt, 768, 1, lds_x, lds_h, node0, N, tid, wave, row, hi);
  encode_feature(nump,   Wn, bn,  16, 2, lds_x, lds_h, node0, N, tid, wave, row, hi);
  encode_feature(cat,    Wc, bc,   8, 3, lds_x, lds_h, node0, N, tid, wave, row, hi);
  __syncthreads();  // make every wave's lds_h slice visible

  // stage 2: nf = relu(h[32,256] @ Wf[256,64] + bf)
  const int col = 16 * wave + row;
  v8f acc0 = {}, acc1 = {};
  for (int k = 0; k < 256; k += 4) {
    v2f b;
    b.x = Wf[(size_t)(k + 2 * hi) * HD + col];
    b.y = Wf[(size_t)(k + 2 * hi + 1) * HD + col];
    v2f a0 = *(const v2f*)(lds_h + row * HSTRIDE + k + 2 * hi);
    v2f a1 = *(const v2f*)(lds_h + (16 + row) * HSTRIDE + k + 2 * hi);
    acc0 = wmma4(a0, b, acc0);
    acc1 = wmma4(a1, b, acc1);
  }
  const float bias = bf[col];
  for (int r8 = 0; r8 < 8; ++r8) {
    int nrow = node0 + r8 + 8 * hi;
    if (nrow < N) nf[(size_t)nrow * HD + col] = fmaxf(acc0[r8] + bias, 0.f);
    nrow += 16;
    if (nrow < N) nf[(size_t)nrow * HD + col] = fmaxf(acc1[r8] + bias, 0.f);
  }
}

// ---------------------------------------------------------------------------
// Kernel 2: edge pipeline. 32 edges per block, 128 threads = 4 waves.
// raw = concat(nf[src], nf[dst]) async-gathered to LDS; emb = relu(raw+Eet);
// m1 = relu(emb@Wm1+bm1) via WMMA; logit = m1@Wm2+bm2; s = (u < sigmoid);
// masked atomic scatter of the raw dst half.
// ---------------------------------------------------------------------------
__global__ __launch_bounds__(128) void edge_kernel(
    const int* __restrict__ ei, const int* __restrict__ et,
    const float* __restrict__ u, const float* __restrict__ nf,
    const float* __restrict__ Eet,
    const float* __restrict__ Wm1, const float* __restrict__ bm1,
    const float* __restrict__ Wm2, const float* __restrict__ bm2,
    float* __restrict__ sums, float* __restrict__ cnt, int E) {
  __shared__ int s_src[32], s_dst[32], s_et[32];
  __shared__ float s_u[32];
  __shared__ float s_raw[32 * ESTRIDE];  // raw concat rows (dst half reused)
  __shared__ float s_emb[32 * ESTRIDE];
  __shared__ float s_m1[32 * MSTRIDE];
  __shared__ float s_part[32 * 4];
  __shared__ float s_sel[32];

  const int tid = threadIdx.x;
  const int e0 = blockIdx.x * 32;
  if (tid < 32) {
    const int e = e0 + tid;
    const bool ok = e < E;
    s_src[tid] = ok ? ei[e] : 0;
    s_dst[tid] = ok ? ei[(size_t)E + e] : 0;
    s_et[tid] = ok ? et[e] : 0;
    s_u[tid] = ok ? u[e] : 2.0f;  // 2.0 > sigmoid(x) -> never selected
  }
  __syncthreads();

  // async gather of nf[src]|nf[dst] rows (32 edges x 32 float4 chunks)
  for (int idx = tid; idx < 32 * 32; idx += 128) {
    const int e = idx >> 5;
    const int c = (idx & 31) * 4;  // float offset within the 128-col row
    const float* g = (c < HD) ? (nf + (size_t)s_src[e] * HD + c)
                              : (nf + (size_t)s_dst[e] * HD + (c - HD));
    async_copy_b128(g, s_raw + e * ESTRIDE + c);
  }
  async_join();
  __syncthreads();

  // emb = relu(raw + Eet[edge_type])
  for (int idx = tid; idx < 32 * 32; idx += 128) {
    const int e = idx >> 5;
    const int c = (idx & 31) * 4;
    float4 v = *(const float4*)(s_raw + e * ESTRIDE + c);
    float4 t4 = *(const float4*)(Eet + (size_t)s_et[e] * 128 + c);
    float4 r;
    r.x = fmaxf(v.x + t4.x, 0.f);
    r.y = fmaxf(v.y + t4.y, 0.f);
    r.z = fmaxf(v.z + t4.z, 0.f);
    r.w = fmaxf(v.w + t4.w, 0.f);
    *(float4*)(s_emb + e * ESTRIDE + c) = r;
  }
  __syncthreads();

  // m1 = relu(emb[32,128] @ Wm1[128,64] + bm1) via WMMA f32 16x16x4
  const int wave = tid >> 5;
  const int lane = tid & 31;
  const int row = lane & 15;
  const int hi = lane >> 4;
  const int col = 16 * wave + row;
  v8f acc0 = {}, acc1 = {};
  for (int k = 0; k < 128; k += 4) {
    v2f b;
    b.x = Wm1[(size_t)(k + 2 * hi) * HD + col];
    b.y = Wm1[(size_t)(k + 2 * hi + 1) * HD + col];
    v2f a0 = *(const v2f*)(s_emb + row * ESTRIDE + k + 2 * hi);
    v2f a1 = *(const v2f*)(s_emb + (16 + row) * ESTRIDE + k + 2 * hi);
    acc0 = wmma4(a0, b, acc0);
    acc1 = wmma4(a1, b, acc1);
  }
  const float bias = bm1[col];
  for (int r8 = 0; r8 < 8; ++r8) {
    s_m1[(r8 + 8 * hi) * MSTRIDE + col] = fmaxf(acc0[r8] + bias, 0.f);
    s_m1[(16 + r8 + 8 * hi) * MSTRIDE + col] = fmaxf(acc1[r8] + bias, 0.f);
  }
  __syncthreads();

  // logit partials: thread (e, sg) reduces 16 columns
  {
    const int e = tid >> 2;
    const int sg = tid & 3;
    float p = 0.f;
    for (int j = 0; j < 16; ++j)
      p = fmaf(s_m1[e * MSTRIDE + sg * 16 + j], Wm2[sg * 16 + j], p);
    s_part[e * 4 + sg] = p;
  }
  __syncthreads();
  if (tid < 32) {
    const float d = bm2[0] + s_part[tid * 4 + 0] + s_part[tid * 4 + 1] +
                    s_part[tid * 4 + 2] + s_part[tid * 4 + 3];
    const float pij = 1.f / (1.f + expf(-d));
    s_sel[tid] = (s_u[tid] < pij) ? 1.f : 0.f;
  }
  __syncthreads();

  // masked scatter: sums[src] += u_j (raw dst half) ; cnt[src] += 1
  for (int pass = 0; pass < 2; ++pass) {
    const int e = pass * 16 + (tid >> 3);
    const int seg = tid & 7;
    if (s_sel[e] > 0.f) {
      float* dp = sums + (size_t)s_src[e] * HD + seg * 8;
      const float* sp = s_raw + e * ESTRIDE + HD + seg * 8;
      for (int j = 0; j < 8; ++j) atomicAdd(dp + j, sp[j]);
      if (seg == 0) atomicAdd(cnt + s_src[e], 1.0f);
    }
  }
}

// ---------------------------------------------------------------------------
// Kernel 3: agg = cnt>0 ? sums/max(cnt,1) : nf ;  out = agg @ Wcls + bcls
// ---------------------------------------------------------------------------
__global__ __launch_bounds__(256) void finalize_kernel(
    const float* __restrict__ sums, const float* __restrict__ cnt,
    const float* __restrict__ nf, const float* __restrict__ Wcls,
    const float* __restrict__ bcls, float* __restrict__ out, int N) {
  const int n = blockIdx.x * 256 + threadIdx.x;
  if (n >= N) return;
  const float c = cnt[n];
  const bool have = (c > 0.f);
  const float* src = have ? (sums + (size_t)n * HD) : (nf + (size_t)n * HD);
  const float scale = have ? (1.f / fmaxf(c, 1.f)) : 1.f;
  float o0 = bcls[0], o1 = bcls[1];
  for (int k = 0; k < HD; ++k) {
    const float a = src[k] * scale;
    o0 = fmaf(a, Wcls[2 * k + 0], o0);
    o1 = fmaf(a, Wcls[2 * k + 1], o1);
  }
  out[2 * n + 0] = o0;
  out[2 * n + 1] = o1;
}

// ---------------------------------------------------------------------------
extern "C" void kernel_launch(void* const* d_in, const int* in_sizes, int n_in,
                              void* d_out, int out_size, void* d_ws,
                              size_t ws_size, hipStream_t stream) {
  const float* image = (const float*)d_in[0];
  const float* tweets = (const float*)d_in[1];
  const float* nump = (const float*)d_in[2];
  const float* cat = (const float*)d_in[3];
  const int* ei = (const int*)d_in[4];
  const int* et = (const int*)d_in[5];
  // d_in[6] = eps : unused (z is dead in the reference output)
  const float* u = (const float*)d_in[7];
  const float* Wi = (const float*)d_in[8];
  const float* bi = (const float*)d_in[9];
  const float* Wt = (const float*)d_in[10];
  const float* bt = (const float*)d_in[11];
  const float* Wn = (const float*)d_in[12];
  const float* bn = (const float*)d_in[13];
  const float* Wc = (const float*)d_in[14];
  const float* bc = (const float*)d_in[15];
  const float* Wf = (const float*)d_in[16];
  const float* bf = (const float*)d_in[17];
  const float* Eet = (const float*)d_in[18];
  // d_in[19..22] = Wmu,bmu,Wlv,blv : unused (mu/log_var dead)
  const float* Wm1 = (const float*)d_in[23];
  const float* bm1 = (const float*)d_in[24];
  const float* Wm2 = (const float*)d_in[25];
  const float* bm2 = (const float*)d_in[26];
  const float* Wcls = (const float*)d_in[27];
  const float* bcls = (const float*)d_in[28];

  const int N = in_sizes[0] / 512;  // 50000
  const int E = in_sizes[5];        // 1000000

  // workspace: nf [N,64] | sums [N,64] | cnt [N]   (~25.8 MB)
  float* nf = (float*)d_ws;
  float* sums = nf + (size_t)N * HD;
  float* cnt = sums + (size_t)N * HD;
  // sums and cnt are contiguous -> single capture-safe async memset
  hipMemsetAsync(sums, 0, ((size_t)N * HD + N) * sizeof(float), stream);

  node_encode_kernel<<<(N + 31) / 32, 128, 0, stream>>>(
      image, tweets, nump, cat, Wi, bi, Wt, bt, Wn, bn, Wc, bc, Wf, bf, nf, N);

  edge_kernel<<<(E + 31) / 32, 128, 0, stream>>>(ei, et, u, nf, Eet, Wm1, bm1,
                                                 Wm2, bm2, sums, cnt, E);

  finalize_kernel<<<(N + 255) / 256, 256, 0, stream>>>(sums, cnt, nf, Wcls,
                                                       bcls, (float*)d_out, N);
}